// MyBertAttnBPWordPiece_20555713478673
// MI455X (gfx1250) — compile-verified
//
#include <hip/hip_runtime.h>

// CDNA5 / gfx1250, wave32. fp32 WMMA (V_WMMA_F32_16X16X4_F32) GEMM pipeline
// with GLOBAL_LOAD_ASYNC_TO_LDS_B128 tile staging (ASYNCcnt path).

typedef __attribute__((ext_vector_type(2))) float v2f;
typedef __attribute__((ext_vector_type(8))) float v8f;

#define Bsz    32
#define Ssz    512
#define Hdim   768
#define Wsz    256
#define Lsz    8
#define NINT   26
#define NSLOT  121
#define MROWS  (Bsz * Wsz)        // 8192

__device__ __forceinline__ int clipS(int s) {
    return s < 0 ? 0 : (s > (Ssz - 1) ? (Ssz - 1) : s);
}

// Direct global -> LDS async copy, 16B per lane. LDS dest address is the low
// 32 bits of the generic pointer (ISA 10.2).
__device__ __forceinline__ void async_copy_b128(void* lds_dst, const void* gsrc) {
    uint32_t lds_off = (uint32_t)(uintptr_t)lds_dst;
    asm volatile("global_load_async_to_lds_b128 %0, %1, off"
                 :: "v"(lds_off), "v"(gsrc)
                 : "memory");
}
__device__ __forceinline__ void wait_asynccnt0() {
    asm volatile("s_wait_asynccnt 0" ::: "memory");
}

// ---------------------------------------------------------------------------
// res_id = res @ lin_id_w^T + lin_id_b   (32 x 26, trivial)
// ---------------------------------------------------------------------------
__global__ void res_id_kernel(const float* __restrict__ res,
                              const float* __restrict__ w,
                              const float* __restrict__ b,
                              float* __restrict__ out) {
    int i = blockIdx.x * blockDim.x + threadIdx.x;
    if (i >= Bsz * NINT) return;
    int bb = i / NINT, o = i % NINT;
    const float* r  = res + (long)bb * Hdim;
    const float* wr = w   + (long)o  * Hdim;
    float acc = b[o];
    #pragma unroll 4
    for (int h = 0; h < Hdim; ++h) acc += r[h] * wr[h];
    out[i] = acc;
}

// ---------------------------------------------------------------------------
// WMMA fp32 GEMM: C[M x N] = Arows @ B  (+ epilogue)
//   WT=true : B[k][n] = Wm[n*K + k]   (weight stored [N x K], "x @ W^T")
//   WT=false: B[k][n] = Wm[k*N + n]   (weight stored [K x N], "x @ W")
//   GATHER  : A row m = res_all[b, clip(tok[m]), :]  (b = m / Wsz), lda = Hdim
//   EPI==1  : word epilogue: val = (len==1) ? first : (acc + bias + first)
// Tile: 128x64, BK=16, 8 waves/block, 4 x v8f accumulators per wave.
// LDS rows padded to 20 floats: 80B stride keeps async B128 stores 16B-aligned
// and makes the b64 fragment reads bank-conflict-free (gcd(20,64)=4, 16
// distinct bank-pairs over 16 lanes).
// ---------------------------------------------------------------------------
template <int EPI, bool WT, bool GATHER>
__global__ __launch_bounds__(256) void gemm_wmma_f32(
    const float* __restrict__ A, const float* __restrict__ Wm,
    const float* __restrict__ bias, float* __restrict__ C,
    int M, int N, int K, int ldc,
    const int* __restrict__ tok, const int* __restrict__ lens,
    const float* __restrict__ res_all) {
    constexpr int BM = 128, BN = 64, BK = 16, BKP = BK + 4;
    __shared__ float As[BM][BKP];
    __shared__ float Bs[BN][BKP];   // B^T tile: Bs[n][k]

    const int tid   = threadIdx.x;
    const int wave  = tid >> 5;
    const int lane  = tid & 31;
    const int mbase = blockIdx.x * BM;
    const int nbase = blockIdx.y * BN;

    v8f acc[4] = {};

    for (int kc = 0; kc < K; kc += BK) {
        __syncthreads();   // previous iteration's fragment reads done
        // ---- A tile (128 x 16): async global->LDS, 16B/lane, 2 passes ----
        {
            const int kq = (tid & 3) * 4;
            #pragma unroll
            for (int p = 0; p < 2; ++p) {
                const int row = (tid >> 2) + p * 64;
                long aoff;
                if constexpr (GATHER) {
                    const int m = mbase + row;
                    const int bb = m / Wsz;
                    const int s = clipS(tok[m]);
                    aoff = (long)(bb * Ssz + s) * Hdim;
                } else {
                    aoff = (long)(mbase + row) * K;
                }
                async_copy_b128(&As[row][kq], A + aoff + kc + kq);
            }
        }
        // ---- B tile ----
        if constexpr (WT) {
            const int n  = tid >> 2;
            const int kq = (tid & 3) * 4;
            int gn = nbase + n;
            if (gn > N - 1) gn = N - 1;   // clamp: cols >= N discarded later
            async_copy_b128(&Bs[n][kq], Wm + (long)gn * K + kc + kq);
        } else {
            // transpose-on-stage (N always a multiple of BN on this path)
            const int k  = tid >> 4;
            const int n4 = (tid & 15) * 4;
            const float4 wv = *(const float4*)(Wm + (long)(kc + k) * N + nbase + n4);
            Bs[n4 + 0][k] = wv.x; Bs[n4 + 1][k] = wv.y;
            Bs[n4 + 2][k] = wv.z; Bs[n4 + 3][k] = wv.w;
        }
        wait_asynccnt0();
        __syncthreads();
        // ---- 16x16x4 f32 WMMA: lanes 0-15 carry K={k,k+1}, 16-31 K={k+2,k+3}
        const int lh = lane & 15;
        const int kh = (lane >> 4) * 2;
        #pragma unroll
        for (int ks = 0; ks < BK; ks += 4) {
            const v2f af = *(const v2f*)&As[wave * 16 + lh][ks + kh];
            #pragma unroll
            for (int nt = 0; nt < 4; ++nt) {
                const v2f bf = *(const v2f*)&Bs[nt * 16 + lh][ks + kh];
                acc[nt] = __builtin_amdgcn_wmma_f32_16x16x4_f32(
                    false, af, false, bf, (short)0, acc[nt], false, false);
            }
        }
    }

    // ---- epilogue: C/D layout VGPR v, lane<16 -> M=v,N=lane; lane>=16 -> M=v+8
    const int lh = lane & 15;
    const int mh = (lane >> 4) * 8;
    #pragma unroll
    for (int nt = 0; nt < 4; ++nt) {
        const int col = nbase + nt * 16 + lh;
        if (col >= N) continue;
        const float bv = bias ? bias[col] : 0.0f;
        #pragma unroll
        for (int v = 0; v < 8; ++v) {
            const int row = mbase + wave * 16 + mh + v;
            float val = acc[nt][v] + bv;
            if constexpr (EPI == 1) {
                const int len = lens[row];
                const int s   = clipS(tok[row]);
                const int bb  = row / Wsz;
                const float first = res_all[(long)(bb * Ssz + s) * Hdim + col];
                val = (len == 1) ? first : (val + first);
            }
            C[(long)row * ldc + col] = val;
        }
    }
}

// ---------------------------------------------------------------------------
// Per-(b,w) wave32 kernel: scores_k = t.g_k + q0.Wsk_b, masked softmax over
// L=8, gbar = sum_k attn_k * g_k.  One wave per row, 8 waves per block.
// ---------------------------------------------------------------------------
__global__ __launch_bounds__(256) void softmax_gbar_kernel(
    const float* __restrict__ res_all, const int* __restrict__ tok,
    const int* __restrict__ lens, const float* __restrict__ t,
    const float* __restrict__ q0, const float* __restrict__ wsk_b,
    float* __restrict__ gbar) {
    const int wid  = (blockIdx.x * blockDim.x + threadIdx.x) >> 5;  // (b,w) row
    const int lane = threadIdx.x & 31;
    if (wid >= MROWS) return;
    const int bb    = wid / Wsz;
    const int start = tok[wid];
    const int len   = lens[wid];

    const float* gp[Lsz];
    #pragma unroll
    for (int k = 0; k < Lsz; ++k)
        gp[k] = res_all + (long)(bb * Ssz + clipS(start + k)) * Hdim;

    const float* trow = t  + (long)wid * Hdim;
    const float* qrow = q0 + (long)wid * Hdim;

    float dot[Lsz] = {};
    float qb = 0.f;
    for (int c = lane; c < Hdim; c += 32) {
        const float tv = trow[c];
        qb += qrow[c] * wsk_b[c];
        #pragma unroll
        for (int k = 0; k < Lsz; ++k) dot[k] += tv * gp[k][c];
    }
    #pragma unroll
    for (int off = 16; off > 0; off >>= 1) {
        #pragma unroll
        for (int k = 0; k < Lsz; ++k) dot[k] += __shfl_xor(dot[k], off, 32);
        qb += __shfl_xor(qb, off, 32);
    }

    float sc[Lsz], mx = -1.0e30f;
    #pragma unroll
    for (int k = 0; k < Lsz; ++k) {
        sc[k] = (k <= len) ? (dot[k] + qb) : -1.0e9f;
        mx = fmaxf(mx, sc[k]);
    }
    float sum = 0.f, a[Lsz];
    #pragma unroll
    for (int k = 0; k < Lsz; ++k) { a[k] = __expf(sc[k] - mx); sum += a[k]; }
    const float inv = 1.0f / sum;
    #pragma unroll
    for (int k = 0; k < Lsz; ++k) a[k] *= inv;

    float* grow = gbar + (long)wid * Hdim;
    for (int c = lane; c < Hdim; c += 32) {
        float g = 0.f;
        #pragma unroll
        for (int k = 0; k < Lsz; ++k) g += a[k] * gp[k][c];
        grow[c] = g;
    }
}

// ---------------------------------------------------------------------------
extern "C" void kernel_launch(void* const* d_in, const int* in_sizes, int n_in,
                              void* d_out, int out_size, void* d_ws, size_t ws_size,
                              hipStream_t stream) {
    const float* res_all    = (const float*)d_in[0];
    const float* res        = (const float*)d_in[1];
    const int*   tok        = (const int*)  d_in[2];
    const int*   lens       = (const int*)  d_in[3];
    const float* Wsq_w      = (const float*)d_in[4];
    const float* Wsq_b      = (const float*)d_in[5];
    const float* Wsk_w      = (const float*)d_in[6];
    const float* Wsk_b      = (const float*)d_in[7];
    const float* Wsv_w      = (const float*)d_in[8];
    const float* Wsv_b      = (const float*)d_in[9];
    const float* lin_id_w   = (const float*)d_in[10];
    const float* lin_id_b   = (const float*)d_in[11];
    const float* lin_slot_w = (const float*)d_in[12];
    const float* lin_slot_b = (const float*)d_in[13];

    float* out_id = (float*)d_out;                 // [32*26]
    float* out_sf = out_id + Bsz * NINT;           // [8192*121]

    const size_t MH = (size_t)MROWS * Hdim;        // 6.29M floats / 24 MB
    float* q0   = (float*)d_ws;                    // stage 2 out
    float* tbuf = q0 + MH;                         // stage 3 out
    float* gbar = tbuf + MH;                       // stage 4 out
    float* word = q0;                              // stage 5 out (q0 dead)

    // 1) res_id (tiny)
    res_id_kernel<<<(Bsz * NINT + 255) / 256, 256, 0, stream>>>(
        res, lin_id_w, lin_id_b, out_id);

    // 2) q0 = gather(g0) @ Wsq^T + bq
    gemm_wmma_f32<0, true, true><<<dim3(MROWS / 128, Hdim / 64), 256, 0, stream>>>(
        res_all, Wsq_w, Wsq_b, q0, MROWS, Hdim, Hdim, Hdim, tok, nullptr, nullptr);

    // 3) t = q0 @ Wsk   (no bias; bias effect handled as q0.Wsk_b scalar)
    gemm_wmma_f32<0, false, false><<<dim3(MROWS / 128, Hdim / 64), 256, 0, stream>>>(
        q0, Wsk_w, nullptr, tbuf, MROWS, Hdim, Hdim, Hdim, nullptr, nullptr, nullptr);

    // 4) masked softmax over L, gbar = sum_k attn_k g_k
    softmax_gbar_kernel<<<MROWS / 8, 256, 0, stream>>>(
        res_all, tok, lens, tbuf, q0, Wsk_b, gbar);

    // 5) word = select(len==1, first, gbar @ Wsv^T + bv + first)
    gemm_wmma_f32<1, true, false><<<dim3(MROWS / 128, Hdim / 64), 256, 0, stream>>>(
        gbar, Wsv_w, Wsv_b, word, MROWS, Hdim, Hdim, Hdim, tok, lens, res_all);

    // 6) res_sf = word @ lin_slot^T + b   (N=121, guarded)
    gemm_wmma_f32<0, true, false><<<dim3(MROWS / 128, 2), 256, 0, stream>>>(
        word, lin_slot_w, lin_slot_b, out_sf, MROWS, NSLOT, Hdim, NSLOT,
        nullptr, nullptr, nullptr);
}